// TFMPTF_Optimized_12171937316944
// MI455X (gfx1250) — compile-verified
//
#include <hip/hip_runtime.h>
#include <hip/hip_bf16.h>

typedef __attribute__((ext_vector_type(2))) float v2f;
typedef __attribute__((ext_vector_type(8))) float v8f;

#define TT   4096
#define BB   16
#define DD   64
#define KK   5
#define DT   8          // channels (d) per workgroup; one wave32 per channel
#define XSTRIDE 4100    // padded T stride in LDS
#define NTAP 17
#define RR   8

// Branch-free ordinal-pattern id of a length-3 window (stable argsort ascending,
// Lehmer code with FACT=[1,1,2]); ranks -> packed LUT.
__device__ __forceinline__ int perm3(float a, float b, float c) {
    int p01 = (a <= b);      // index 0 precedes 1 (ties: lower index first)
    int p02 = (a <= c);
    int p12 = (b <= c);
    int r0 = 2 - p01 - p02;  // rank of element 0
    int r1 = p01 + 1 - p12;  // rank of element 1
    int key = r0 * 3 + r1;   // (r2 implied)
    return (int)((0xAE0440u >> (3 * key)) & 7u);
}

__global__ __launch_bounds__(256)
void tfmptf_fused_kernel(const float* __restrict__ x, float* __restrict__ out)
{
    __shared__ float xs[DT * XSTRIDE];       // channel time series; reused as symmask scratch
    __shared__ float modebuf[DT][KK][64];    // 64-slot mode ring per channel
    __shared__ float Gs[KK][NTAP];           // FIR taps (exact DFT of symmetrized mask)
    __shared__ int   counts[DT][36];         // 6x6 transition histogram per channel
    __shared__ float S1buf[DT][KK];          // sum of energies
    __shared__ float covbuf[DT][25];         // raw Gram (from WMMA)

    const int tid  = threadIdx.x;
    const int lane = tid & 31;
    const int wv   = tid >> 5;               // wave id == local channel
    const int b    = blockIdx.x >> 3;
    const int d0   = (blockIdx.x & 7) * DT;

    // ---- zero init ----
    if (tid < KK * NTAP) (&Gs[0][0])[tid] = 0.0f;
    for (int i = tid; i < DT * 36; i += 256) (&counts[0][0])[i] = 0;
    if (tid < DT * KK) (&S1buf[0][0])[tid] = 0.0f;
    __syncthreads();

    // ---- symmetrized Gaussian masks into xs-scratch: m~(f) = (m(f)+m(-f))/2 ----
    float* symmask = xs;                     // [KK][TT]
    for (int idx = tid; idx < KK * TT; idx += 256) {
        int k = idx >> 12, f = idx & (TT - 1);
        float fr = (f < TT / 2) ? f * (1.0f / TT) : (f - TT) * (1.0f / TT);
        float c  = 0.2f * k - 0.5f;          // centers (k - K/2)/K
        float d1 = fr - c, d2 = -fr - c;     // bandwidth 1/K -> exp(-12.5*(fr-c)^2)
        symmask[idx] = 0.5f * (expf(-12.5f * d1 * d1) + expf(-12.5f * d2 * d2));
    }
    __syncthreads();

    // ---- exact time-domain taps: g_k[tau] = (1/T) sum_f m~_k[f] cos(2*pi*f*tau/T) ----
    for (int task = tid; task < KK * NTAP * 16; task += 256) {
        int tap = task >> 4, ch = task & 15;
        int k = tap / NTAP, j = tap % NTAP, tau = j - RR;
        float part = 0.0f;
        int f0 = ch * (TT / 16);
        for (int f = f0; f < f0 + TT / 16; ++f) {
            int ph = ((f * tau) % TT + TT) % TT;
            part += symmask[k * TT + f] *
                    cosf((float)ph * (6.28318530717958647692f / TT));
        }
        atomicAdd(&Gs[k][j], part * (1.0f / TT));
    }
    __syncthreads();

    // ---- load x[b, :, d0..d0+7] transposed into LDS (overwrites mask scratch) ----
    {
        const float* hb = x + ((size_t)b * TT * DD) + d0;
        int trow = tid >> 2;
        int pr   = (tid & 3) * 2;
        for (int tb = 0; tb < TT; tb += 64) {
            int t = tb + trow;
            const float* p = hb + (size_t)t * DD + pr;
            float a0 = p[0], a1 = p[1];
            xs[pr * XSTRIDE + t]       = a0;
            xs[(pr + 1) * XSTRIDE + t] = a1;
        }
    }
    __syncthreads();

    // ---- fused streaming pass: modes (17-tap circular FIR), ordinal transitions,
    //      energy sums, and WMMA Gram accumulation ----
    const float* xrow = &xs[wv * XSTRIDE];
    float s1[KK] = {0, 0, 0, 0, 0};
    v8f acc = {0, 0, 0, 0, 0, 0, 0, 0};
    const int row = lane & 15;               // WMMA A/B row (k index, zero-padded to 16)
    const int kb  = (lane >> 4) * 2;         // K-slot base per half-wave (ISA 16x4 layout)

    for (int c = 0; c <= TT / 32; ++c) {
        if (c < TT / 32) {                   // produce 32 new mode samples per channel
            int t = c * 32 + lane;
            float mk[KK] = {0, 0, 0, 0, 0};
            #pragma unroll
            for (int j = 0; j < NTAP; ++j) {
                float xv = xrow[(t + j - RR) & (TT - 1)];
                #pragma unroll
                for (int k = 0; k < KK; ++k) mk[k] += Gs[k][j] * xv;
            }
            #pragma unroll
            for (int k = 0; k < KK; ++k) {
                modebuf[wv][k][t & 63] = mk[k];
                s1[k] += mk[k] * mk[k];      // sum of energies e = m^2
            }
        }
        __syncthreads();

        if (c >= 1) {                        // ordinal transitions of previous chunk
            int w = (c - 1) * 32 + lane;
            if (w <= TT - 4) {               // transitions exist for w in [0, W-2]=[0,4092]
                #pragma unroll
                for (int k = 0; k < KK; ++k) {
                    float v0 = modebuf[wv][k][w & 63];
                    float v1 = modebuf[wv][k][(w + 1) & 63];
                    float v2 = modebuf[wv][k][(w + 2) & 63];
                    float v3 = modebuf[wv][k][(w + 3) & 63];
                    int idc = perm3(v0, v1, v2);
                    int idn = perm3(v1, v2, v3);
                    atomicAdd(&counts[wv][idc * 6 + idn], 1);
                }
            }
        }

        if (c < TT / 32) {                   // Gram rank-4 updates on current chunk
            #pragma unroll
            for (int u = 0; u < 8; ++u) {
                int ta = c * 32 + u * 4 + kb;
                float e0 = 0.0f, e1 = 0.0f;
                if (row < KK) {
                    float x0 = modebuf[wv][row][ta & 63];
                    float x1 = modebuf[wv][row][(ta + 1) & 63];
                    e0 = x0 * x0;
                    e1 = x1 * x1;
                }
                v2f ab = {e0, e1};
                // D = A*B + C, A = E[:,t..t+3] (16x4), B = A^T (4x16): identical
                // per-lane register contents per the ISA 32-bit A/B layouts.
                acc = __builtin_amdgcn_wmma_f32_16x16x4_f32(
                        false, ab, false, ab, (short)0, acc, false, false);
            }
        }
        __syncthreads();
    }

    // ---- reductions: energy sums + Gram extraction (D: lanes 0-15 hold N, vgpr r = M) ----
    #pragma unroll
    for (int k = 0; k < KK; ++k) atomicAdd(&S1buf[wv][k], s1[k]);
    if (lane < KK) {
        #pragma unroll
        for (int r = 0; r < KK; ++r) covbuf[wv][r * KK + lane] = acc[r];
    }
    __syncthreads();

    // ---- features: 36 transition probs + 10 correlations per channel ----
    const int iu[10] = {0, 0, 0, 0, 1, 1, 1, 2, 2, 3};
    const int ju[10] = {1, 2, 3, 4, 2, 3, 4, 3, 4, 4};
    for (int oi = tid; oi < DT * 46; oi += 256) {
        int dl = oi / 46, f = oi % 46;
        float val;
        if (f < 36) {
            int cr = f / 6;
            float rs = 0.0f;
            for (int n = 0; n < 6; ++n) rs += (float)counts[dl][cr * 6 + n];
            val = (float)counts[dl][f] / (rs == 0.0f ? 1.0f : rs);
        } else {
            int p = f - 36;
            int i = iu[p], j = ju[p];
            float mi  = S1buf[dl][i] * (1.0f / TT);
            float mj  = S1buf[dl][j] * (1.0f / TT);
            float cij = covbuf[dl][i * KK + j] - (float)TT * mi * mj;
            float cii = covbuf[dl][i * KK + i] - (float)TT * mi * mi;
            float cjj = covbuf[dl][j * KK + j] - (float)TT * mj * mj;
            float den = sqrtf(fmaxf(cii, 0.0f)) * sqrtf(fmaxf(cjj, 0.0f));
            val = (den > 0.0f) ? (cij / den) : 0.0f;
            val = fminf(1.0f, fmaxf(-1.0f, val));
        }
        out[((size_t)b * DD + (size_t)(d0 + dl)) * 46 + f] = val;
    }
}

extern "C" void kernel_launch(void* const* d_in, const int* in_sizes, int n_in,
                              void* d_out, int out_size, void* d_ws, size_t ws_size,
                              hipStream_t stream) {
    (void)in_sizes; (void)n_in; (void)d_ws; (void)ws_size; (void)out_size;
    const float* xin = (const float*)d_in[0];
    float* outp = (float*)d_out;
    dim3 grid(BB * (DD / DT));   // 128 workgroups: 16 batches x 8 channel-groups
    dim3 block(256);             // 8 wave32 waves, one per channel
    tfmptf_fused_kernel<<<grid, block, 0, stream>>>(xin, outp);
}